// MambaBlock_74191265071864
// MI455X (gfx1250) — compile-verified
//
#include <hip/hip_runtime.h>
#include <cstdint>

#define D_MODEL 1024
#define D_STATE 16
#define D_CONV  4
#define D_INNER 2048
#define DT_RANK 64
#define BATCH   4
#define SEQLEN  2048
#define NROWS   (BATCH * SEQLEN)   // 8192 token rows

typedef __attribute__((ext_vector_type(2))) float v2f;
typedef __attribute__((ext_vector_type(8))) float v8f;

__device__ __forceinline__ int imin(int a, int b) { return a < b ? a : b; }
__device__ __forceinline__ float siluf(float x) { return x / (1.0f + expf(-x)); }
__device__ __forceinline__ float softplusf(float x) {
    // numerically stable log1p(exp(x))
    return fmaxf(x, 0.0f) + log1pf(expf(-fabsf(x)));
}

// ---------------------------------------------------------------------------
// RMSNorm: one block per token row of D_MODEL=1024, 256 threads * float4.
// ---------------------------------------------------------------------------
__global__ __launch_bounds__(256) void rmsnorm_kernel(const float* __restrict__ x,
                                                      const float* __restrict__ w,
                                                      float* __restrict__ out) {
    const int row = blockIdx.x;
    const int tid = threadIdx.x;
    const float* xr = x + (size_t)row * D_MODEL;
    float4 v = *(const float4*)(xr + tid * 4);
    float ss = v.x * v.x + v.y * v.y + v.z * v.z + v.w * v.w;
    __shared__ float red[256];
    red[tid] = ss;
    __syncthreads();
    for (int s = 128; s > 0; s >>= 1) {
        if (tid < s) red[tid] += red[tid + s];
        __syncthreads();
    }
    const float inv = 1.0f / sqrtf(red[0] * (1.0f / D_MODEL) + 1e-5f);
    const float4 wv = *(const float4*)(w + tid * 4);
    float4 ov;
    ov.x = v.x * inv * wv.x;
    ov.y = v.y * inv * wv.y;
    ov.z = v.z * inv * wv.z;
    ov.w = v.w * inv * wv.w;
    *(float4*)(out + (size_t)row * D_MODEL + tid * 4) = ov;
}

// ---------------------------------------------------------------------------
// fp32 WMMA GEMM: C[M,N] = act(A[M,K(lda)] @ W[N,K]^T + bias) + resid
// Wave computes a 64x64 tile via 16 v_wmma_f32_16x16x4_f32 per K-step.
// Block = 8 waves -> 128x256 tile. N handled with clamped loads + masked
// stores (WMMA requires EXEC all-1s; control flow stays wave-uniform).
// ---------------------------------------------------------------------------
__global__ __launch_bounds__(256) void wmma_gemm_kernel(
    const float* __restrict__ A, int lda,
    const float* __restrict__ W,        // N x K, row-major
    const float* __restrict__ bias,     // N or nullptr
    const float* __restrict__ resid,    // M x N or nullptr
    float* __restrict__ C,              // M x N
    int M, int N, int K, int act)       // act: 0=none, 1=softplus
{
    const int tid  = threadIdx.x;
    const int lane = tid & 31;
    const int wave = tid >> 5;       // 0..7
    const int wm   = wave >> 2;      // 0..1
    const int wn   = wave & 3;       // 0..3
    const int m0   = blockIdx.y * 128 + wm * 64;
    const int n0   = blockIdx.x * 256 + wn * 64;
    const int r    = lane & 15;           // row within 16-tile (A:m, B:n)
    const int kl   = (lane >> 4) << 1;    // 0 or 2: per-lane K sub-offset

    // Per-lane row base pointers (clamped for OOB tiles; stores are masked).
    const float* arow[4];
    const float* brow[4];
#pragma unroll
    for (int i = 0; i < 4; i++)
        arow[i] = A + (size_t)imin(m0 + i * 16 + r, M - 1) * lda + kl;
#pragma unroll
    for (int j = 0; j < 4; j++)
        brow[j] = W + (size_t)imin(n0 + j * 16 + r, N - 1) * K + kl;

    v8f acc[16];
#pragma unroll
    for (int i = 0; i < 16; i++)
#pragma unroll
        for (int e = 0; e < 8; e++) acc[i][e] = 0.0f;

    for (int k = 0; k < K; k += 4) {
        v2f af[4], bf[4];
#pragma unroll
        for (int i = 0; i < 4; i++) af[i] = *(const v2f*)(arow[i] + k);
#pragma unroll
        for (int j = 0; j < 4; j++) bf[j] = *(const v2f*)(brow[j] + k);
#pragma unroll
        for (int i = 0; i < 4; i++)
#pragma unroll
            for (int j = 0; j < 4; j++)
                acc[i * 4 + j] = __builtin_amdgcn_wmma_f32_16x16x4_f32(
                    false, af[i], false, bf[j], (short)0, acc[i * 4 + j],
                    false, false);
    }

    // C/D layout: VGPR e -> M = e + 8*(lane>=16); N = lane&15.
    const int col = lane & 15;
    const int rof = (lane >> 4) * 8;
#pragma unroll
    for (int i = 0; i < 4; i++) {
#pragma unroll
        for (int j = 0; j < 4; j++) {
            const int gn = n0 + j * 16 + col;
            if (gn >= N) continue;
            v8f av = acc[i * 4 + j];
#pragma unroll
            for (int e = 0; e < 8; e++) {
                const int gm = m0 + i * 16 + rof + e;
                if (gm >= M) continue;
                float val = av[e];
                if (bias)  val += bias[gn];
                if (act)   val  = softplusf(val);
                if (resid) val += resid[(size_t)gm * N + gn];
                C[(size_t)gm * N + gn] = val;
            }
        }
    }
}

// ---------------------------------------------------------------------------
// Depthwise causal conv (K=4) + SiLU. Reads xi = xz[..., :2048] (stride 4096),
// writes xc (B,L,2048).
// ---------------------------------------------------------------------------
__global__ __launch_bounds__(256) void conv_silu_kernel(const float* __restrict__ xz,
                                                        const float* __restrict__ cw,
                                                        const float* __restrict__ cb,
                                                        float* __restrict__ xc) {
    const size_t idx = (size_t)blockIdx.x * 256 + threadIdx.x;
    if (idx >= (size_t)NROWS * D_INNER) return;
    const int    d    = (int)(idx & (D_INNER - 1));
    const size_t row  = idx >> 11;              // token row (b*L + l)
    const int    l    = (int)(row & (SEQLEN - 1));
    const size_t brow = row - (size_t)l;        // b*L
    float acc = cb[d];
#pragma unroll
    for (int k = 0; k < D_CONV; k++) {
        const int ls = l - (D_CONV - 1) + k;
        if (ls >= 0)
            acc += xz[(brow + (size_t)ls) * (2 * D_INNER) + d] * cw[d * D_CONV + k];
    }
    xc[idx] = siluf(acc);
}

// ---------------------------------------------------------------------------
// Selective scan, sequential over L, parallel over (B, D_INNER).
// Per thread: one channel d, 16 recurrent states in registers.
// B_t/C_t (32 floats/step, shared by all d) staged into LDS in 64-step chunks.
// Fuses u*D skip term and silu(z) gate into the output.
// ---------------------------------------------------------------------------
#define SCAN_CHUNK 64
__global__ __launch_bounds__(256) void scan_kernel(
    const float* __restrict__ xc,     // (B,L,2048) u
    const float* __restrict__ dtb,    // (B,L,2048) softplus(dt)
    const float* __restrict__ xdbl,   // (B,L,96): B at +64, C at +80
    const float* __restrict__ A_log,  // (2048,16)
    const float* __restrict__ Dv,     // (2048)
    const float* __restrict__ xz,     // (B,L,4096): z at +2048
    float* __restrict__ y)            // (B,L,2048)
{
    __shared__ float sm[SCAN_CHUNK * 32];
    const int    tid = threadIdx.x;
    const int    b   = blockIdx.x >> 3;
    const int    d   = ((blockIdx.x & 7) << 8) + tid;
    const size_t bL  = (size_t)b * SEQLEN;

    float a[D_STATE], h[D_STATE];
#pragma unroll
    for (int n = 0; n < D_STATE; n++) {
        a[n] = -expf(A_log[d * D_STATE + n]);
        h[n] = 0.0f;
    }
    const float Dd = Dv[d];

    for (int l0 = 0; l0 < SEQLEN; l0 += SCAN_CHUNK) {
        __syncthreads();   // previous chunk fully consumed
        for (int i = tid; i < SCAN_CHUNK * 32; i += 256) {
            const int t = i >> 5, c = i & 31;
            sm[i] = xdbl[(bL + (size_t)(l0 + t)) * 96 + 64 + c];
        }
        __syncthreads();
        for (int t = 0; t < SCAN_CHUNK; t++) {
            const size_t ridx = bL + (size_t)(l0 + t);
            const float dtv = dtb[ridx * D_INNER + d];
            const float u   = xc [ridx * D_INNER + d];
            const float zz  = xz [ridx * (2 * D_INNER) + D_INNER + d];
            const float dtu = dtv * u;
            const float* Bp = &sm[t * 32];
            float acc = 0.0f;
#pragma unroll
            for (int n = 0; n < D_STATE; n++) {
                const float dA = expf(dtv * a[n]);
                h[n] = dA * h[n] + dtu * Bp[n];
                acc += h[n] * Bp[16 + n];
            }
            acc += u * Dd;
            acc *= siluf(zz);
            y[ridx * D_INNER + d] = acc;
        }
    }
}

// ---------------------------------------------------------------------------
// Launch: 2 layers. Param order per layer (flattened dict order):
// norm_w, in_proj, conv_w, conv_b, x_proj, dt_proj_w, dt_proj_b, A_log, D, out_proj
// ---------------------------------------------------------------------------
extern "C" void kernel_launch(void* const* d_in, const int* in_sizes, int n_in,
                              void* d_out, int out_size, void* d_ws, size_t ws_size,
                              hipStream_t stream) {
    (void)in_sizes; (void)n_in; (void)out_size; (void)ws_size;
    const float* x0  = (const float*)d_in[0];
    float*       out = (float*)d_out;
    float*       ws  = (float*)d_ws;

    size_t off = 0;
    float* xn   = ws + off; off += (size_t)NROWS * D_MODEL;        // 8M
    float* xz   = ws + off; off += (size_t)NROWS * 2 * D_INNER;    // 32M
    float* xc   = ws + off; off += (size_t)NROWS * D_INNER;        // 16M
    float* xdbl = ws + off; off += (size_t)NROWS * 96;             // 0.75M
    float* dtb  = ws + off; off += (size_t)NROWS * D_INNER;        // 16M
    float* yb   = ws + off; off += (size_t)NROWS * D_INNER;        // 16M

    for (int layer = 0; layer < 2; ++layer) {
        const float* const* p = (const float* const*)(d_in + 1 + layer * 10);
        const float* norm_w  = p[0];
        const float* in_proj = p[1];
        const float* conv_w  = p[2];
        const float* conv_b  = p[3];
        const float* x_proj  = p[4];
        const float* dt_w    = p[5];
        const float* dt_b    = p[6];
        const float* A_log   = p[7];
        const float* Dvec    = p[8];
        const float* out_w   = p[9];
        const float* xin = (layer == 0) ? x0 : out;

        // 1) xn = rmsnorm(x)
        rmsnorm_kernel<<<NROWS, 256, 0, stream>>>(xin, norm_w, xn);

        // 2) xz = xn @ in_proj^T   (8192 x 4096 x 1024)
        {
            dim3 g((2 * D_INNER + 255) / 256, (NROWS + 127) / 128);
            wmma_gemm_kernel<<<g, 256, 0, stream>>>(xn, D_MODEL, in_proj, nullptr,
                                                    nullptr, xz, NROWS, 2 * D_INNER,
                                                    D_MODEL, 0);
        }

        // 3) xc = silu(causal_conv(xi))
        {
            const size_t total = (size_t)NROWS * D_INNER;
            conv_silu_kernel<<<(unsigned)((total + 255) / 256), 256, 0, stream>>>(
                xz, conv_w, conv_b, xc);
        }

        // 4) xdbl = xc @ x_proj^T   (8192 x 96 x 2048)
        {
            dim3 g((96 + 255) / 256, (NROWS + 127) / 128);
            wmma_gemm_kernel<<<g, 256, 0, stream>>>(xc, D_INNER, x_proj, nullptr,
                                                    nullptr, xdbl, NROWS, 96,
                                                    D_INNER, 0);
        }

        // 5) dt = softplus(dt_r @ dt_proj_w^T + dt_bias)  (8192 x 2048 x 64)
        {
            dim3 g((D_INNER + 255) / 256, (NROWS + 127) / 128);
            wmma_gemm_kernel<<<g, 256, 0, stream>>>(xdbl, 96, dt_w, dt_b, nullptr,
                                                    dtb, NROWS, D_INNER, DT_RANK, 1);
        }

        // 6) y = selective_scan(...) * silu(z)  (gate + skip fused)
        scan_kernel<<<BATCH * (D_INNER / 256), 256, 0, stream>>>(
            xc, dtb, xdbl, A_log, Dvec, xz, yb);

        // 7) x_out = residual + y @ out_proj^T  (8192 x 1024 x 2048)
        {
            dim3 g((D_MODEL + 255) / 256, (NROWS + 127) / 128);
            wmma_gemm_kernel<<<g, 256, 0, stream>>>(yb, D_INNER, out_w, nullptr,
                                                    xin, out, NROWS, D_MODEL,
                                                    D_INNER, 0);
        }
    }
}